// CFAR_OS_2D_75849122448295
// MI455X (gfx1250) — compile-verified
//
#include <hip/hip_runtime.h>
#include <math.h>
#include <stdint.h>

#define V_DIM 512
#define R_DIM 1024
#define TILE  32
#define HALO  6
#define LW    44          /* 32 + 2*6 */
#define LSTRIDE 45
#define RANK0 36          /* N - K = 144 - 108 */

// ---------------------------------------------------------------------------
// Kernel: OS-CFAR 2D. One block computes a 32x32 output tile.
// Stage 44x44 halo tile into LDS via CDNA5 async global->LDS loads, then each
// thread radix-selects the 36th largest of its 144 ring cells (MSB-first,
// 4 bits/pass, exact on IEEE bit patterns since all inputs are >= 0).
// ---------------------------------------------------------------------------
__global__ __launch_bounds__(256)
void cfar_os_2d_kernel(const float* __restrict__ data,
                       float* __restrict__ out, float alpha) {
  __shared__ float tile[LW * LSTRIDE];

  const int tid = threadIdx.x;
  const int V0 = blockIdx.y * TILE;
  const int R0 = blockIdx.x * TILE;

  const uint64_t sbase   = (uint64_t)(uintptr_t)data;          // uniform -> SGPR pair
  const unsigned ldsBase = (unsigned)(uintptr_t)(&tile[0]);    // low 32b = LDS byte offset

  // ---- async stage: global (wrapped) -> LDS -------------------------------
  for (int e = tid; e < LW * LW; e += 256) {
    int r = e / LW;
    int c = e - r * LW;
    int gr = (V0 - HALO + r) & (V_DIM - 1);
    int gc = (R0 - HALO + c) & (R_DIM - 1);
    unsigned goff  = ((unsigned)((gr << 10) + gc)) << 2;       // byte offset into data
    unsigned laddr = ldsBase + (unsigned)(r * LSTRIDE + c) * 4u;
    asm volatile("global_load_async_to_lds_b32 %0, %1, %2"
                 :: "v"(laddr), "v"(goff), "s"(sbase) : "memory");
  }
  asm volatile("s_wait_asynccnt 0" ::: "memory");
  __syncthreads();

  const int tx  = tid & (TILE - 1);   // column within tile (consecutive per wave)
  const int ty0 = tid >> 5;           // 0..7 ; row is uniform within a wave

  #pragma unroll 1
  for (int p = 0; p < 4; ++p) {
    const int ty = ty0 + (p << 3);

    // Radix select: find bit pattern of the 36th largest ring value.
    unsigned prefix = 0u;
    int rank = RANK0;

    #pragma unroll 1
    for (int shift = 28; shift >= 0; shift -= 4) {
      const unsigned maskHi = ~(0xFFFFFFFFu >> (28 - shift)); // bits above current nibble
      unsigned h0 = 0u, h1 = 0u, h2 = 0u, h3 = 0u;            // 16 bins x 8-bit packed

      #pragma unroll 1
      for (int wy = 0; wy < 13; ++wy) {
        const bool gy = ((unsigned)(wy - 4)) <= 4u;           // rows 4..8 hold guard cols
        const float* rowp = &tile[(ty + wy) * LSTRIDE + tx];
        #pragma unroll
        for (int wx = 0; wx < 13; ++wx) {
          const bool gx = (wx >= 4 && wx <= 8);               // compile-time per wx
          unsigned b = __float_as_uint(rowp[wx]);
          bool match = (((b ^ prefix) & maskHi) == 0u);
          if (gx) match = match && !gy;                       // exclude guard hole
          unsigned nib = (b >> shift) & 15u;
          unsigned add = match ? (1u << ((nib & 3u) << 3)) : 0u;
          unsigned sel = nib >> 2;
          h0 += (sel == 0u) ? add : 0u;
          h1 += (sel == 1u) ? add : 0u;
          h2 += (sel == 2u) ? add : 0u;
          h3 += (sel == 3u) ? add : 0u;
        }
      }

      // Walk bins from the top until cumulative count reaches `rank`.
      int r = rank;
      unsigned chosen = 0u;
      bool done = false;
      #pragma unroll
      for (int j = 15; j >= 0; --j) {
        unsigned hv = (j >= 12) ? h3 : (j >= 8) ? h2 : (j >= 4) ? h1 : h0;
        int c = (int)((hv >> ((j & 3) << 3)) & 0xFFu);
        bool take = (!done) && (r <= c);
        chosen = take ? (unsigned)j : chosen;
        done = done || take;
        r = done ? r : (r - c);
      }
      rank = r;
      prefix |= chosen << shift;
    }

    out[(V0 + ty) * R_DIM + (R0 + tx)] = alpha * __uint_as_float(prefix);
  }
}

// ---------------------------------------------------------------------------
// Host-side faithful port of the reference threshold solver (double precision,
// deterministic; runs on CPU inside kernel_launch).
// ---------------------------------------------------------------------------
static double log_factorial_ref(int m) {
  double n = (double)m + 1.0;            // note the reference's n+1 shift
  if (n < 9.0) {
    double f = 1.0;
    for (int i = 2; i <= m + 1; ++i) f *= (double)i;
    return log(f);
  }
  return 0.5 * (log(2.0 * M_PI) - log(n))
       + n * (log(n + 1.0 / (12.0 * n - 1.0 / (10.0 * n))) - 1.0);
}

static double fun_ref(int k, int n, double t, double pfa) {
  double s = 0.0;
  for (int i = n; i > n - k; --i) s += log((double)i + t);
  return log_factorial_ref(n) - log_factorial_ref(n - k) - s - log(pfa);
}

static float compute_alpha() {
  double lo = 1.0, hi = 1e32;
  for (int it = 0; it < 300; ++it) {
    double mid = 0.5 * (lo + hi);
    if (fun_ref(108, 144, mid, 1e-5) > 0.0) lo = mid; else hi = mid;
  }
  return (float)(0.5 * (lo + hi));
}

extern "C" void kernel_launch(void* const* d_in, const int* in_sizes, int n_in,
                              void* d_out, int out_size, void* d_ws, size_t ws_size,
                              hipStream_t stream) {
  (void)in_sizes; (void)n_in; (void)out_size; (void)d_ws; (void)ws_size;
  const float* data = (const float*)d_in[0];
  float* out = (float*)d_out;
  const float alpha = compute_alpha();     // deterministic host-side scalar

  dim3 grid(R_DIM / TILE, V_DIM / TILE);   // 32 x 16 blocks
  cfar_os_2d_kernel<<<grid, 256, 0, stream>>>(data, out, alpha);
}